// YOLOLoss_23905787970056
// MI455X (gfx1250) — compile-verified
//
#include <hip/hip_runtime.h>
#include <math.h>

// ---------------- problem constants ----------------
#define BATCH 16
#define MTGT  60
#define ANCH  3
#define NCLS  80

// workspace layout (bytes)
#define MASK_TOTAL 134400                      // 16*6400 + 16*1600 + 16*400
#define REC_OFF    134400                      // 960 records * 48 B
#define OBJP_OFF   (REC_OFF + BATCH*MTGT*48)   // 180480
#define OBJ_BLOCKS 512
#define OBJ_SPLIT0 390
#define OBJ_SPLIT1 488
#define TGTP_OFF   (OBJP_OFF + OBJ_BLOCKS*3*4) // 186624
// total ws needed: TGTP_OFF + 960*4*4 = 201984 bytes

__device__ __forceinline__ int scaleW(int s)   { return s == 0 ? 80 : (s == 1 ? 40 : 20); }
__device__ __forceinline__ int scaleHW(int s)  { return s == 0 ? 6400 : (s == 1 ? 1600 : 400); }
__device__ __forceinline__ int maskOffB(int s) { return s == 0 ? 0 : (s == 1 ? 102400 : 128000); }

__device__ __forceinline__ float softplusf(float x) {
  // stable log(1+exp(x)) == max(x,0) + log1p(exp(-|x|))  (matches jax.nn.softplus)
  return fmaxf(x, 0.0f) + log1pf(expf(-fabsf(x)));
}
__device__ __forceinline__ float sigmoidf(float x) {
  return 1.0f / (1.0f + expf(-x));
}

// ---------------- wave32 reduction via V_WMMA_F32_16X16X4_F32 ----------------
typedef __attribute__((ext_vector_type(2))) float v2f;
typedef __attribute__((ext_vector_type(8))) float v8f;

__device__ __forceinline__ float wave_sum(float v) {
#if defined(__gfx1250__) && __has_builtin(__builtin_amdgcn_wmma_f32_16x16x4_f32)
  // A(16x4): VGPR0 carries one distinct A element per lane (our value), VGPR1 = 0.
  // B(4x16) = all ones -> layout-independent: D[m,n] = rowsum(A[m,:]).
  v2f a;    a[0] = v;    a[1] = 0.0f;
  v2f ones; ones[0] = 1.0f; ones[1] = 1.0f;
  v8f c = {0.f,0.f,0.f,0.f,0.f,0.f,0.f,0.f};
  v8f d = __builtin_amdgcn_wmma_f32_16x16x4_f32(false, a, false, ones,
                                                (short)0, c, false, false);
  // D layout: VGPR i, lanes 0-15 -> row i; lanes 16-31 -> row i+8.
  float t = ((d[0] + d[1]) + (d[2] + d[3])) + ((d[4] + d[5]) + (d[6] + d[7]));
  // lanes<16 hold sum(rows 0..7), lanes>=16 hold sum(rows 8..15); combine:
  return __shfl(t, 0, 32) + __shfl(t, 16, 32);
#else
  for (int off = 16; off > 0; off >>= 1) v += __shfl_xor(v, off, 32);
  return v;
#endif
}

// Valid on threadIdx.x == 0. All threads must call (contains barriers).
__device__ __forceinline__ float block_sum(float v, float* smem, int nwaves) {
  float w = wave_sum(v);
  if ((threadIdx.x & 31) == 0) smem[threadIdx.x >> 5] = w;
  __syncthreads();
  float r = 0.0f;
  if (threadIdx.x == 0)
    for (int i = 0; i < nwaves; ++i) r += smem[i];
  __syncthreads();
  return r;
}

// ---------------- kernel 1: per-target metadata + pos2d masks ----------------
__global__ void build_targets_kernel(const float* __restrict__ boxes,
                                     const int* __restrict__ labels,
                                     unsigned char* __restrict__ ws) {
  int t = blockIdx.x * blockDim.x + threadIdx.x;
  if (t >= BATCH * MTGT) return;
  int b = t / MTGT;
  const float* bx = boxes + (size_t)t * 4;
  float x1 = bx[0], y1 = bx[1], x2 = bx[2], y2 = bx[3];
  float bw = fminf(fmaxf((x2 - x1) / 640.0f, 1e-6f), 1.0f);
  float bh = fminf(fmaxf((y2 - y1) / 640.0f, 1e-6f), 1.0f);
  float cx = fminf(fmaxf((x1 + x2) * 0.5f / 640.0f, 0.0f), 1.0f - 1e-6f);
  float cy = fminf(fmaxf((y1 + y2) * 0.5f / 640.0f, 0.0f), 1.0f - 1e-6f);
  float ms = fmaxf(bw, bh);
  int s = ms < 0.15f ? 0 : (ms < 0.45f ? 1 : 2);
  int lab = labels[t];
  int valid = (lab >= 0 && lab < NCLS) ? 1 : 0;
  int W = scaleW(s);
  int gi = (int)floorf(cx * (float)W); gi = min(max(gi, 0), W - 1);
  int gj = (int)floorf(cy * (float)W); gj = min(max(gj, 0), W - 1);  // H == W

  int* ri = (int*)(ws + REC_OFF + (size_t)t * 48);
  float* rf = (float*)(ri + 4);
  ri[0] = valid; ri[1] = s; ri[2] = gi; ri[3] = gj;
  rf[0] = cx - bw * 0.5f; rf[1] = cy - bh * 0.5f;
  rf[2] = cx + bw * 0.5f; rf[3] = cy + bh * 0.5f;
  ri[8] = lab; ri[9] = 0; ri[10] = 0; ri[11] = 0;

  if (valid) {
    int hw = scaleHW(s);
    ws[maskOffB(s) + b * hw + gj * W + gi] = 1;  // same-value racy store: OK
  }
}

// ---------------- kernel 2: objectness focal loss over channel-4 planes ------
__global__ void obj_focal_kernel(const float* __restrict__ p3,
                                 const float* __restrict__ p4,
                                 const float* __restrict__ p5,
                                 const unsigned char* __restrict__ ws,
                                 float* __restrict__ objPart) {
  __shared__ float smem[8];
  int bid = blockIdx.x;
  int s, bls, nbls;
  if (bid < OBJ_SPLIT0)      { s = 0; bls = bid;              nbls = OBJ_SPLIT0; }
  else if (bid < OBJ_SPLIT1) { s = 1; bls = bid - OBJ_SPLIT0; nbls = OBJ_SPLIT1 - OBJ_SPLIT0; }
  else                       { s = 2; bls = bid - OBJ_SPLIT1; nbls = OBJ_BLOCKS - OBJ_SPLIT1; }
  const float* pred = (s == 0) ? p3 : (s == 1) ? p4 : p5;
  const unsigned char* mask = ws + maskOffB(s);
  int hw = scaleHW(s);
  int N = BATCH * ANCH * hw;
  int stride = nbls * (int)blockDim.x;
  float pos = 0.0f, neg = 0.0f, cnt = 0.0f;
  for (int i = bls * (int)blockDim.x + (int)threadIdx.x; i < N; i += stride) {
    int pix = i % hw;
    int plane = i / hw;
    int a = plane % ANCH;
    int b = plane / ANCH;
    float x = pred[(size_t)((b * 255 + a * 85 + 4) * hw + pix)];
    int m = mask[b * hw + pix];
    float p = sigmoidf(x);
    if (m) {   // t = 1: bce = softplus(-x), a_t = 0.25, (1-p_t) = 1-p
      float om = 1.0f - p;
      pos += softplusf(-x) * (0.25f * om * om);
      cnt += 1.0f;
    } else {   // t = 0: bce = softplus(x), a_t = 0.75, (1-p_t) = p
      neg += softplusf(x) * (0.75f * p * p);
    }
  }
  float rp = block_sum(pos, smem, 8);
  float rn = block_sum(neg, smem, 8);
  float rc = block_sum(cnt, smem, 8);
  if (threadIdx.x == 0) {
    objPart[bid * 3 + 0] = rp;
    objPart[bid * 3 + 1] = rn;
    objPart[bid * 3 + 2] = rc;
  }
}

// ---------------- kernel 3: per-target cls (deduped per cell) + box GIoU -----
__global__ void pos_cls_box_kernel(const float* __restrict__ p3,
                                   const float* __restrict__ p4,
                                   const float* __restrict__ p5,
                                   const unsigned char* __restrict__ ws,
                                   float* __restrict__ tgtPart) {
  __shared__ unsigned int clsMask[3];
  __shared__ int sOwner, sValid, sScale, sGi, sGj;
  __shared__ float sT[4];
  __shared__ float smem[2];
  int t = blockIdx.x;
  int b = t / MTGT, m = t % MTGT;

  if (threadIdx.x == 0) {
    const int* ri = (const int*)(ws + REC_OFF + (size_t)t * 48);
    const float* rf = (const float*)(ri + 4);
    int valid = ri[0], s = ri[1], gi = ri[2], gj = ri[3];
    sValid = valid; sScale = s; sGi = gi; sGj = gj;
    sT[0] = rf[0]; sT[1] = rf[1]; sT[2] = rf[2]; sT[3] = rf[3];
    unsigned int cm0 = 0, cm1 = 0, cm2 = 0;
    int owner = valid;
    if (valid) {
      const unsigned char* recs = ws + REC_OFF + (size_t)b * MTGT * 48;
      for (int j = 0; j < MTGT; ++j) {
        const int* rj = (const int*)(recs + (size_t)j * 48);
        if (rj[0] && rj[1] == s && rj[2] == gi && rj[3] == gj) {
          if (j < m) owner = 0;          // an earlier target owns this cell
          int lj = rj[8];
          if (lj < 32)      cm0 |= 1u << lj;
          else if (lj < 64) cm1 |= 1u << (lj - 32);
          else              cm2 |= 1u << (lj - 64);
        }
      }
    }
    sOwner = owner;
    clsMask[0] = cm0; clsMask[1] = cm1; clsMask[2] = cm2;
  }
  __syncthreads();

  int valid = sValid;
  int s = sScale, gi = sGi, gj = sGj;
  const float* pred = (s == 0) ? p3 : (s == 1) ? p4 : p5;
  int W = scaleW(s), hw = scaleHW(s);
  int cellOff = gj * W + gi;
  float clsLocal = 0.0f, boxLocal = 0.0f;

  if (valid) {
    if (sOwner) {
      for (int idx = (int)threadIdx.x; idx < ANCH * NCLS; idx += (int)blockDim.x) {
        int a = idx / NCLS;
        int c = idx % NCLS;
        float x = pred[(size_t)((b * 255 + a * 85 + 5 + c) * hw + cellOff)];
        unsigned int bit = (clsMask[c >> 5] >> (c & 31)) & 1u;
        clsLocal += bit ? softplusf(-x) : softplusf(x);
      }
    }
    if (threadIdx.x < ANCH) {
      int a = (int)threadIdx.x;
      float v0 = pred[(size_t)((b * 255 + a * 85 + 0) * hw + cellOff)];
      float v1 = pred[(size_t)((b * 255 + a * 85 + 1) * hw + cellOff)];
      float v2 = pred[(size_t)((b * 255 + a * 85 + 2) * hw + cellOff)];
      float v3 = pred[(size_t)((b * 255 + a * 85 + 3) * hw + cellOff)];
      float sx = sigmoidf(v0), sy = sigmoidf(v1);
      float pw = sigmoidf(v2), ph = sigmoidf(v3);
      float pcx = ((float)gi + sx) / (float)W;
      float pcy = ((float)gj + sy) / (float)W;   // h == w
      float px1 = pcx - pw * 0.5f, py1 = pcy - ph * 0.5f;
      float px2 = pcx + pw * 0.5f, py2 = pcy + ph * 0.5f;
      float tx1 = sT[0], ty1 = sT[1], tx2 = sT[2], ty2 = sT[3];
      float a1 = (px2 - px1) * (py2 - py1);
      float a2 = (tx2 - tx1) * (ty2 - ty1);
      float iw = fmaxf(fminf(px2, tx2) - fmaxf(px1, tx1), 0.0f);
      float ih = fmaxf(fminf(py2, ty2) - fmaxf(py1, ty1), 0.0f);
      float inter = iw * ih;
      float uni = a1 + a2 - inter;
      float iou = inter / uni;
      float ew = fmaxf(fmaxf(px2, tx2) - fminf(px1, tx1), 0.0f);
      float eh = fmaxf(fmaxf(py2, ty2) - fminf(py1, ty1), 0.0f);
      float enc = ew * eh;
      float g = iou - (enc - uni) / enc;
      boxLocal = 1.0f - g;
    }
  }

  float rc = block_sum(clsLocal, smem, 2);
  float rb = block_sum(boxLocal, smem, 2);
  if (threadIdx.x == 0) {
    tgtPart[t * 4 + 0] = rc;
    tgtPart[t * 4 + 1] = rb;
    tgtPart[t * 4 + 2] = valid ? 1.0f : 0.0f;
    tgtPart[t * 4 + 3] = (float)s;
  }
}

// ---------------- kernel 4: deterministic final combine ----------------------
__global__ void finalize_kernel(const float* __restrict__ objPart,
                                const float* __restrict__ tgtPart,
                                float* __restrict__ out) {
  __shared__ float smem[8];
  float pos[3] = {0, 0, 0}, neg[3] = {0, 0, 0}, cnt[3] = {0, 0, 0};
  for (int i = (int)threadIdx.x; i < OBJ_BLOCKS; i += (int)blockDim.x) {
    int s = (i < OBJ_SPLIT0) ? 0 : (i < OBJ_SPLIT1) ? 1 : 2;
    float p = objPart[i * 3 + 0], n = objPart[i * 3 + 1], c = objPart[i * 3 + 2];
    if (s == 0)      { pos[0] += p; neg[0] += n; cnt[0] += c; }
    else if (s == 1) { pos[1] += p; neg[1] += n; cnt[1] += c; }
    else             { pos[2] += p; neg[2] += n; cnt[2] += c; }
  }
  float cls[3] = {0, 0, 0};
  float box = 0.0f, nval = 0.0f;
  for (int i = (int)threadIdx.x; i < BATCH * MTGT; i += (int)blockDim.x) {
    float c = tgtPart[i * 4 + 0];
    float bx = tgtPart[i * 4 + 1];
    float v = tgtPart[i * 4 + 2];
    int s = (int)tgtPart[i * 4 + 3];
    if (s == 0) cls[0] += c; else if (s == 1) cls[1] += c; else cls[2] += c;
    box += bx; nval += v;
  }
  float red[14];
  red[0]  = block_sum(pos[0], smem, 8);
  red[1]  = block_sum(pos[1], smem, 8);
  red[2]  = block_sum(pos[2], smem, 8);
  red[3]  = block_sum(neg[0], smem, 8);
  red[4]  = block_sum(neg[1], smem, 8);
  red[5]  = block_sum(neg[2], smem, 8);
  red[6]  = block_sum(cnt[0], smem, 8);
  red[7]  = block_sum(cnt[1], smem, 8);
  red[8]  = block_sum(cnt[2], smem, 8);
  red[9]  = block_sum(box,    smem, 8);
  red[10] = block_sum(nval,   smem, 8);
  red[11] = block_sum(cls[0], smem, 8);
  red[12] = block_sum(cls[1], smem, 8);
  red[13] = block_sum(cls[2], smem, 8);
  if (threadIdx.x == 0) {
    const float L_NOOBJ = 1.0f, L_OBJ = 1.0f, L_CLS = 0.5f, L_BOX = 5.0f;
    float obj_loss = 0.0f, cls_loss = 0.0f;
    for (int s = 0; s < 3; ++s) {
      float c = red[6 + s];
      obj_loss += (L_OBJ * red[s] + L_NOOBJ * red[3 + s]) / fmaxf(c, 1.0f);
      cls_loss += red[11 + s] / fmaxf(c * (float)NCLS, 1.0f);
    }
    float total_pos = fmaxf(red[10] * (float)ANCH, 1.0f);
    out[0] = obj_loss / 3.0f + L_CLS * (cls_loss / 3.0f) + L_BOX * red[9] / total_pos;
  }
}

// ---------------- host launcher ----------------------------------------------
extern "C" void kernel_launch(void* const* d_in, const int* in_sizes, int n_in,
                              void* d_out, int out_size, void* d_ws, size_t ws_size,
                              hipStream_t stream) {
  const float* p3     = (const float*)d_in[0];
  const float* p4     = (const float*)d_in[1];
  const float* p5     = (const float*)d_in[2];
  const float* boxes  = (const float*)d_in[3];
  const int*   labels = (const int*)d_in[4];
  unsigned char* ws = (unsigned char*)d_ws;
  float* objPart = (float*)(ws + OBJP_OFF);
  float* tgtPart = (float*)(ws + TGTP_OFF);

  // pos2d masks must start at zero every call (records/partials are fully
  // overwritten each call, so only the mask region needs clearing).
  hipMemsetAsync(ws, 0, MASK_TOTAL, stream);

  build_targets_kernel<<<(BATCH * MTGT + 255) / 256, 256, 0, stream>>>(boxes, labels, ws);
  obj_focal_kernel<<<OBJ_BLOCKS, 256, 0, stream>>>(p3, p4, p5, ws, objPart);
  pos_cls_box_kernel<<<BATCH * MTGT, 64, 0, stream>>>(p3, p4, p5, ws, tgtPart);
  finalize_kernel<<<1, 256, 0, stream>>>(objPart, tgtPart, (float*)d_out);
}